// LaplacianPyramid_2396591751864
// MI455X (gfx1250) — compile-verified
//
#include <hip/hip_runtime.h>
#include <hip/hip_bf16.h>
#include <math.h>

// ---------------------------------------------------------------------------
// NLPD (Normalized Laplacian Pyramid Distance), B=16, C=3, 512x512, K=6.
// Bandwidth-bound (~135MB total traffic vs 23.3TB/s): per level, one LDS-tiled
// stride-2 down-conv per image, plus one fully fused kernel:
//   async J-tile -> LDS (overlapped with bilinear upsample into LDS)
//   -> 5x5 conv + residual -> 3x3 |.| conv -> divisive norm (v_rcp_f32)
//   -> squared diff -> wave reduction via V_WMMA_F32_16X16X4_F32 -> atomicAdd.
// ---------------------------------------------------------------------------

typedef __attribute__((ext_vector_type(2))) float v2f;
typedef __attribute__((ext_vector_type(8))) float v8f;

#if defined(__gfx1250__)
#define NLPD_ASYNC_LDS 1
#else
#define NLPD_ASYNC_LDS 0
#endif

__device__ __forceinline__ int reflect_i(int i, int n) {
    // jnp.pad(..., mode='reflect'): mirror about edge sample (pad <= n-1 here)
    if (i < 0) i = -i;
    if (i >= n) i = 2 * n - 2 - i;
    return i;
}

__device__ __forceinline__ void async_f32_to_lds(void* lds_dst, const float* gptr) {
#if NLPD_ASYNC_LDS
    // Generic LDS pointer: low 32 bits are the workgroup-relative LDS offset
    // (ISA 10.2: LDS aperture lives in addr[63:32]).
    const unsigned loff = (unsigned)(uintptr_t)lds_dst;
    asm volatile("global_load_async_to_lds_b32 %0, %1, off"
                 :: "v"(loff), "v"(gptr) : "memory");
#else
    *(float*)lds_dst = *gptr;
#endif
}

__device__ __forceinline__ void wait_async_lds() {
#if NLPD_ASYNC_LDS
#if __has_builtin(__builtin_amdgcn_s_wait_asynccnt)
    __builtin_amdgcn_s_wait_asynccnt(0);
#else
    asm volatile("s_wait_asynccnt 0x0" ::: "memory");
#endif
#endif
}

// ---------------------------------------------------------------------------
// Depthwise 5x5 stride-2 conv, reflect SAME padding (pt=pl=1, pb=pr=2).
// src: (B*C, H, H) -> dst: (B*C, H/2, H/2). 16x16 output tile per block,
// 35x35 reflect-padded input tile staged through LDS (one global touch/elem).
// ---------------------------------------------------------------------------
__global__ __launch_bounds__(256) void down5_k(const float* __restrict__ src,
                                               float* __restrict__ dst,
                                               const float* __restrict__ filt,
                                               int H) {
    constexpr int IN = 35;
    __shared__ float T[IN][IN + 1];
    __shared__ float Fs[25];

    const int Ho = H >> 1;
    const int z = blockIdx.z;
    const int c = z % 3;
    const int tx = threadIdx.x, ty = threadIdx.y;
    const int tid = ty * 16 + tx;
    if (tid < 25) Fs[tid] = filt[c * 25 + tid];

    const int ox0 = blockIdx.x * 16, oy0 = blockIdx.y * 16;
    const float* sp = src + (size_t)z * H * H;

    // CDNA5 global_prefetch_b8: start pulling a late row of this slice's tile region.
    __builtin_prefetch(sp + (size_t)reflect_i(min(2 * oy0 + 31, H + 1), H) * H, 0, 1);

    // Reflect-padded input tile: rows/cols [2*o?0 - 1, 2*o?0 + 34)
    for (int idx = tid; idx < IN * IN; idx += 256) {
        const int ly = idx / IN, lx = idx % IN;
        const int ry = reflect_i(min(2 * oy0 - 1 + ly, H + 1), H);
        const int rx = reflect_i(min(2 * ox0 - 1 + lx, H + 1), H);
        T[ly][lx] = sp[(size_t)ry * H + rx];
    }
    __syncthreads();

    const int ox = ox0 + tx, oy = oy0 + ty;
    if (ox < Ho && oy < Ho) {
        float acc = 0.f;
#pragma unroll
        for (int ky = 0; ky < 5; ++ky)
#pragma unroll
            for (int kx = 0; kx < 5; ++kx)
                acc = fmaf(Fs[ky * 5 + kx], T[2 * ty + ky][2 * tx + kx], acc);
        dst[(size_t)z * Ho * Ho + (size_t)oy * Ho + ox] = acc;
    }
}

// ---------------------------------------------------------------------------
// Fused per-level kernel, TS x TS output tile per 256-thread block.
//  issue:   async-load J tiles (both images) into LDS       [ASYNCcnt]
//  stage 1: bilinear (align_corners) upsample of I into (TS+6)^2 LDS tile
//  stage 2: out = J - conv5(up) into (TS+2)^2 LDS tile
//  stage 3: dn = conv3(|out|); q = out * rcp(sigma+dn); sum (q1-q2)^2
//  reduce:  per-wave 32->16 via V_WMMA_F32_16X16X4_F32, LDS, shuffle, atomic.
// No early exit before the WMMA -> EXEC is all ones (ISA requirement).
// ---------------------------------------------------------------------------
template <int TS>
__global__ __launch_bounds__(256) void fused_k(const float* __restrict__ J1,
                                               const float* __restrict__ J2,
                                               const float* __restrict__ I1,
                                               const float* __restrict__ I2,
                                               const float* __restrict__ filt,
                                               const float* __restrict__ dnf,
                                               const float* __restrict__ sigmas,
                                               int level,
                                               float* __restrict__ accLevel,
                                               int H) {
    constexpr int UP = TS + 6;   // upsample window, halo 3
    constexpr int OT = TS + 2;   // residual window, halo 1
    const int W = H, h = H >> 1;

    __shared__ float upT[2][UP][UP + 1];
    __shared__ float outT[2][OT][OT + 1];
    __shared__ float JT[2][OT][OT + 1];
    __shared__ float Fs[25];
    __shared__ float Ds[9];
    __shared__ float red[128];

    const int z = blockIdx.z;
    const int c = z % 3;
    const int b = z / 3;
    const int tid = threadIdx.y * 16 + threadIdx.x;

    if (tid < 25) Fs[tid] = filt[c * 25 + tid];
    if (tid >= 32 && tid < 41) Ds[tid - 32] = dnf[c * 9 + (tid - 32)];
    const float sigma = sigmas[level];

    const int tx0 = blockIdx.x * TS, ty0 = blockIdx.y * TS;
    const size_t zJ = (size_t)z * H * W;
    const size_t zI = (size_t)z * h * h;
    const float scale = (float)(h - 1) / (float)(H - 1);

    // ---- issue async J-tile loads (completion overlapped with stage 1) ----
    for (int idx = tid; idx < OT * OT; idx += 256) {
        const int ly = idx / OT, lx = idx % OT;
        const int gy = min(max(ty0 - 1 + ly, 0), H - 1);  // halo entries outside
        const int gx = min(max(tx0 - 1 + lx, 0), W - 1);  // the image are unused
        const size_t p = zJ + (size_t)gy * W + gx;
        async_f32_to_lds(&JT[0][ly][lx], J1 + p);
        async_f32_to_lds(&JT[1][ly][lx], J2 + p);
    }

    // ---- stage 1: bilinear upsample (window [t?0-3, t?0+TS+3)) ----
    for (int idx = tid; idx < UP * UP; idx += 256) {
        const int ly = idx / UP, lx = idx % UP;
        int gy = min(max(ty0 - 3 + ly, 0), H - 1);
        int gx = min(max(tx0 - 3 + lx, 0), W - 1);
        const float fy = gy * scale, fx = gx * scale;
        int y0 = min((int)fy, h - 2);
        int x0 = min((int)fx, h - 2);
        const float wy = fy - (float)y0, wx = fx - (float)x0;
        const size_t p = zI + (size_t)y0 * h + x0;
        {
            const float a = I1[p], bb_ = I1[p + 1], cc = I1[p + h], dd = I1[p + h + 1];
            const float t0 = a + (bb_ - a) * wx, t1 = cc + (dd - cc) * wx;
            upT[0][ly][lx] = t0 + (t1 - t0) * wy;
        }
        {
            const float a = I2[p], bb_ = I2[p + 1], cc = I2[p + h], dd = I2[p + h + 1];
            const float t0 = a + (bb_ - a) * wx, t1 = cc + (dd - cc) * wx;
            upT[1][ly][lx] = t0 + (t1 - t0) * wy;
        }
    }
    wait_async_lds();   // per-wave: our async J loads have landed in LDS
    __syncthreads();    // all waves: upT + JT visible block-wide

    // ---- stage 2: residual out = J - conv5(up), window [t?0-1, t?0+TS+1) ----
    for (int idx = tid; idx < OT * OT; idx += 256) {
        const int ly = idx / OT, lx = idx % OT;
        const int gy = ty0 - 1 + ly, gx = tx0 - 1 + lx;
        float o0 = 0.f, o1 = 0.f;
        if ((unsigned)gy < (unsigned)H && (unsigned)gx < (unsigned)W) {
            float s0 = 0.f, s1 = 0.f;
#pragma unroll
            for (int ky = 0; ky < 5; ++ky) {
                const int ur = reflect_i(gy - 2 + ky, H) - (ty0 - 3);
#pragma unroll
                for (int kx = 0; kx < 5; ++kx) {
                    const int uc = reflect_i(gx - 2 + kx, W) - (tx0 - 3);
                    const float w = Fs[ky * 5 + kx];
                    s0 = fmaf(w, upT[0][ur][uc], s0);
                    s1 = fmaf(w, upT[1][ur][uc], s1);
                }
            }
            o0 = JT[0][ly][lx] - s0;
            o1 = JT[1][ly][lx] - s1;
        }
        outT[0][ly][lx] = o0;
        outT[1][ly][lx] = o1;
    }
    __syncthreads();

    // ---- stage 3: divisive normalization + squared difference ----
    float local = 0.f;
    for (int idx = tid; idx < TS * TS; idx += 256) {   // uniform trip count
        const int py = idx / TS, px = idx % TS;
        const int oy = ty0 + py, ox = tx0 + px;
        float dn0 = 0.f, dn1 = 0.f;
#pragma unroll
        for (int ky = 0; ky < 3; ++ky) {
            const int ur = reflect_i(oy - 1 + ky, H) - (ty0 - 1);
#pragma unroll
            for (int kx = 0; kx < 3; ++kx) {
                const int uc = reflect_i(ox - 1 + kx, W) - (tx0 - 1);
                const float w = Ds[ky * 3 + kx];
                dn0 = fmaf(w, fabsf(outT[0][ur][uc]), dn0);
                dn1 = fmaf(w, fabsf(outT[1][ur][uc]), dn1);
            }
        }
        // v_rcp_f32 (1 ulp) instead of the IEEE div_fixup chain
        const float q0 = outT[0][py + 1][px + 1] * __builtin_amdgcn_rcpf(sigma + dn0);
        const float q1 = outT[1][py + 1][px + 1] * __builtin_amdgcn_rcpf(sigma + dn1);
        const float d = q0 - q1;
        if ((oy < H) && (ox < W)) local = fmaf(d, d, local);
    }

    // ---- wave 32->16 reduction via WMMA (EXEC all ones here) ----
    // A = Ones(16x4); lane n supplies B[0][n] (n<16) / B[2][n-16] (n>=16):
    // every D column n equals local_n + local_{n+16}.
    float S;
#if defined(__gfx1250__) && __has_builtin(__builtin_amdgcn_wmma_f32_16x16x4_f32)
    {
        v2f a;  a[0] = 1.f;   a[1] = 1.f;
        v2f bb; bb[0] = local; bb[1] = 0.f;
        v8f cz = {0.f, 0.f, 0.f, 0.f, 0.f, 0.f, 0.f, 0.f};
        v8f dacc = __builtin_amdgcn_wmma_f32_16x16x4_f32(
            false, a, false, bb, (short)0, cz, false, false);
        S = dacc[0];  // lane n holds local_{n%16} + local_{n%16+16}
    }
#else
    S = local + __shfl_down(local, 16, 32);
#endif
    const int lane = tid & 31, wave = tid >> 5;
    if (lane < 16) red[wave * 16 + lane] = S;
    __syncthreads();
    if (tid < 32) {
        float t = red[tid] + red[tid + 32] + red[tid + 64] + red[tid + 96];
#pragma unroll
        for (int off = 16; off > 0; off >>= 1) t += __shfl_xor(t, off, 32);
        if (tid == 0) atomicAdd(&accLevel[b], t);
    }
}

// ---------------------------------------------------------------------------
// Accumulator init + finalize
// ---------------------------------------------------------------------------
__global__ void zero96_k(float* __restrict__ acc) {
    if (threadIdx.x < 96) acc[threadIdx.x] = 0.f;
}

__global__ void finalize_k(const float* __restrict__ acc, float* __restrict__ out) {
    __shared__ float s[96];
    const int t = threadIdx.x;
    if (t < 96) {
        const int k = t / 16;                 // level
        const int Hk = 512 >> k;              // J size at that level
        const float cnt = 3.0f * (float)Hk * (float)Hk;
        const float tv = sqrtf(acc[t] / cnt); // rmse per (level, batch)
        s[t] = powf(tv, 0.6f);
    }
    __syncthreads();
    if (t == 0) {
        float sum = 0.f;
        for (int i = 0; i < 96; ++i) sum += s[i];
        out[0] = powf(sum, 1.0f / 0.6f);
    }
}

// ---------------------------------------------------------------------------
// Launch: per level, two down-convs then one fused distance kernel.
// 32x32 tiles for H>=32, 16x16 for the 16x16 level. Graph-safe, all on stream.
// ---------------------------------------------------------------------------
extern "C" void kernel_launch(void* const* d_in, const int* in_sizes, int n_in,
                              void* d_out, int out_size, void* d_ws, size_t ws_size,
                              hipStream_t stream) {
    const float* x1     = (const float*)d_in[0];   // (16,3,512,512)
    const float* x2     = (const float*)d_in[1];
    const float* filt   = (const float*)d_in[2];   // (3,1,5,5)
    const float* dnf    = (const float*)d_in[3];   // (6,3,1,3,3)
    const float* sigmas = (const float*)d_in[4];   // (6,)
    float* out = (float*)d_out;

    const int BC = 48;                              // 16 batches * 3 channels
    float* wsf = (float*)d_ws;
    float* acc = wsf;                               // 96 accumulators
    float* buf = wsf + 128;                         // pyramid scratch (aligned)

    // Per-image downsampled level buffers: level i holds (BC, 512>>(i+1))^2
    float* Jb1[6];
    float* Jb2[6];
    size_t cur = 0;
    for (int i = 0; i < 6; ++i) {
        const size_t d = (size_t)(512 >> (i + 1));
        Jb1[i] = buf + cur; cur += (size_t)BC * d * d;
    }
    for (int i = 0; i < 6; ++i) {
        const size_t d = (size_t)(512 >> (i + 1));
        Jb2[i] = buf + cur; cur += (size_t)BC * d * d;
    }
    // cur = 8,386,560 floats (~33.6 MB) of scratch.
    (void)ws_size; (void)in_sizes; (void)n_in; (void)out_size;

    zero96_k<<<1, 128, 0, stream>>>(acc);

    const dim3 blk(16, 16, 1);
    for (int i = 0; i < 6; ++i) {
        const int H = 512 >> i;
        const int Ho = H >> 1;
        const float* J1 = (i == 0) ? x1 : Jb1[i - 1];
        const float* J2 = (i == 0) ? x2 : Jb2[i - 1];

        const dim3 gdown((Ho + 15) / 16, (Ho + 15) / 16, BC);
        down5_k<<<gdown, blk, 0, stream>>>(J1, Jb1[i], filt, H);
        down5_k<<<gdown, blk, 0, stream>>>(J2, Jb2[i], filt, H);

        if (H >= 32) {
            const dim3 gfuse(H / 32, H / 32, BC);
            fused_k<32><<<gfuse, blk, 0, stream>>>(J1, J2, Jb1[i], Jb2[i],
                                                   filt, dnf + (size_t)i * 27,
                                                   sigmas, i, acc + i * 16, H);
        } else {
            const dim3 gfuse(H / 16, H / 16, BC);
            fused_k<16><<<gfuse, blk, 0, stream>>>(J1, J2, Jb1[i], Jb2[i],
                                                   filt, dnf + (size_t)i * 27,
                                                   sigmas, i, acc + i * 16, H);
        }
    }

    finalize_k<<<1, 128, 0, stream>>>(acc, out);
}